// ESN_48447231099060
// MI455X (gfx1250) — compile-verified
//
#include <hip/hip_runtime.h>
#include <hip/hip_bf16.h>

typedef __attribute__((ext_vector_type(2))) float v2f;
typedef __attribute__((ext_vector_type(8))) float v8f;

#define Bsz 64
#define Tsz 256
#define Isz 128
#define Rsz 2048
#define Osz 10
#define ALPHA_ 0.5f

// One wave computes a 32x32 tile (2x2 of 16x16 WMMA tiles, 4 accumulators) of
//   res_t = (1-a)*res_{t-1} + a*tanh(x_t @ i2h + res_{t-1} @ h2h).
// Register tiling amortizes loads: per K=4 step -> 2 x b64 (A) + 4 x b32 (B) for 4 WMMAs
// (1.5 vmem/wmma vs 3.0 for a 16x16 wave tile) and gives 4 independent WMMA chains of ILP.
// res_t lives directly in the hiddens output slab (B,T,R): step t reads slice t-1, writes t.
// 128 waves/step -> 16 blocks x 256 threads. h2h (16MB) stays L2-resident across all steps.
__global__ __launch_bounds__(256)
void esn_step(const float* __restrict__ x,    // (B,T,I)
              const float* __restrict__ i2h,  // (I,R)
              const float* __restrict__ h2h,  // (R,R)
              float* __restrict__ hid,        // (B,T,R)
              int t)
{
    const int wave = threadIdx.x >> 5;
    const int lane = threadIdx.x & 31;
    const int w    = blockIdx.x * 8 + wave;   // 0..127
    const int m0   = (w >> 6) * 32;           // 0 or 32   (batch rows)
    const int n0   = (w & 63) * 32;           // reservoir cols, step 32

    const int half = lane >> 4;               // 0: K+{0,1}, 1: K+{2,3}
    const int l15  = lane & 15;

    v8f acc00 = {}, acc01 = {}, acc10 = {}, acc11 = {};

    // ---- res_{t-1} @ h2h  (wave-uniform skip at t==0: res0 == 0, EXEC stays full) ----
    if (t > 0) {
        const float* a0 = hid + (size_t)(m0 + l15)      * (Tsz * Rsz) + (size_t)(t - 1) * Rsz;
        const float* a1 = hid + (size_t)(m0 + 16 + l15) * (Tsz * Rsz) + (size_t)(t - 1) * Rsz;
        const float* b0 = h2h + (n0 + l15);
        const float* b1 = h2h + (n0 + 16 + l15);
        #pragma unroll 4
        for (int k = 0; k < Rsz; k += 4) {
            const int kk = k + 2 * half;
            v2f aA, aB, bA, bB;
            aA.x = a0[kk];  aA.y = a0[kk + 1];
            aB.x = a1[kk];  aB.y = a1[kk + 1];
            bA.x = b0[(size_t)kk * Rsz];  bA.y = b0[(size_t)(kk + 1) * Rsz];
            bB.x = b1[(size_t)kk * Rsz];  bB.y = b1[(size_t)(kk + 1) * Rsz];
            acc00 = __builtin_amdgcn_wmma_f32_16x16x4_f32(false, aA, false, bA, (short)0, acc00, false, false);
            acc01 = __builtin_amdgcn_wmma_f32_16x16x4_f32(false, aA, false, bB, (short)0, acc01, false, false);
            acc10 = __builtin_amdgcn_wmma_f32_16x16x4_f32(false, aB, false, bA, (short)0, acc10, false, false);
            acc11 = __builtin_amdgcn_wmma_f32_16x16x4_f32(false, aB, false, bB, (short)0, acc11, false, false);
        }
    }

    // ---- x_t @ i2h_w  (K = 128), folded into the same accumulators ----
    {
        const float* a0 = x + (size_t)(m0 + l15)      * (Tsz * Isz) + (size_t)t * Isz;
        const float* a1 = x + (size_t)(m0 + 16 + l15) * (Tsz * Isz) + (size_t)t * Isz;
        const float* b0 = i2h + (n0 + l15);
        const float* b1 = i2h + (n0 + 16 + l15);
        #pragma unroll 4
        for (int k = 0; k < Isz; k += 4) {
            const int kk = k + 2 * half;
            v2f aA, aB, bA, bB;
            aA.x = a0[kk];  aA.y = a0[kk + 1];
            aB.x = a1[kk];  aB.y = a1[kk + 1];
            bA.x = b0[(size_t)kk * Rsz];  bA.y = b0[(size_t)(kk + 1) * Rsz];
            bB.x = b1[(size_t)kk * Rsz];  bB.y = b1[(size_t)(kk + 1) * Rsz];
            acc00 = __builtin_amdgcn_wmma_f32_16x16x4_f32(false, aA, false, bA, (short)0, acc00, false, false);
            acc01 = __builtin_amdgcn_wmma_f32_16x16x4_f32(false, aA, false, bB, (short)0, acc01, false, false);
            acc10 = __builtin_amdgcn_wmma_f32_16x16x4_f32(false, aB, false, bA, (short)0, acc10, false, false);
            acc11 = __builtin_amdgcn_wmma_f32_16x16x4_f32(false, aB, false, bB, (short)0, acc11, false, false);
        }
    }

    // ---- Epilogue: leaky update; C/D layout: VGPR j -> M = j + 8*(lane>=16), N = n0 + (lane&15) ----
    const v8f* accs[4] = { &acc00, &acc01, &acc10, &acc11 };
    #pragma unroll
    for (int q = 0; q < 4; ++q) {
        const int m_base = m0 + (q >> 1) * 16;
        const int n      = n0 + (q & 1) * 16 + l15;
        #pragma unroll
        for (int j = 0; j < 8; ++j) {
            const int m = m_base + j + 8 * half;
            const size_t idx = (size_t)m * (Tsz * Rsz) + (size_t)t * Rsz + n;
            const float prev = (t > 0) ? hid[idx - Rsz] : 0.0f;
            hid[idx] = (1.0f - ALPHA_) * prev + ALPHA_ * tanhf((*accs[q])[j]);
        }
    }
}

// Tiny readout: out[b,o] = res_last[b,:] . out_w[:,o] + out_b[o]   (640 dots of length 2048)
__global__ __launch_bounds__(64)
void esn_readout(const float* __restrict__ hid,  // (B,T,R)
                 const float* __restrict__ ow,   // (R,O)
                 const float* __restrict__ ob,   // (O)
                 float* __restrict__ out)        // (B,O)
{
    const int i = blockIdx.x * blockDim.x + threadIdx.x;
    if (i >= Bsz * Osz) return;
    const int b = i / Osz, o = i % Osz;
    const float* h = hid + (size_t)b * (Tsz * Rsz) + (size_t)(Tsz - 1) * Rsz;
    float s = ob[o];
    #pragma unroll 4
    for (int r = 0; r < Rsz; ++r)
        s += h[r] * ow[(size_t)r * Osz + o];
    out[i] = s;
}

extern "C" void kernel_launch(void* const* d_in, const int* in_sizes, int n_in,
                              void* d_out, int out_size, void* d_ws, size_t ws_size,
                              hipStream_t stream) {
    const float* x   = (const float*)d_in[0];
    const float* i2h = (const float*)d_in[1];
    const float* h2h = (const float*)d_in[2];
    const float* ow  = (const float*)d_in[3];
    const float* ob  = (const float*)d_in[4];

    float* out = (float*)d_out;          // (B,O) = 640 floats
    float* hid = out + Bsz * Osz;        // (B,T,R) follows; 2560B offset keeps alignment

    for (int t = 0; t < Tsz; ++t)
        esn_step<<<16, 256, 0, stream>>>(x, i2h, h2h, hid, t);
    esn_readout<<<10, 64, 0, stream>>>(hid, ow, ob, out);
}